// GraphStackedMultiHeadAttention_11501922419376
// MI455X (gfx1250) — compile-verified
//
#include <hip/hip_runtime.h>
#include <hip/hip_bf16.h>

typedef __attribute__((ext_vector_type(2))) float v2f;
typedef __attribute__((ext_vector_type(8))) float v8f;

#define T_ 8
#define A_ 4
#define N_ 128
#define E_ 2048
#define G_ (T_*A_)
#define NODES (G_*N_)      // 4096
#define EDG (G_*E_)        // 65536
#define RP0 12
#define RP1 4
#define HH 32              // hidden H
#define HEADS 4
#define KK 16

// ---------------------------------------------------------------------------
// Generic fp32 GEMM via V_WMMA_F32_16X16X4_F32.
// Y[r, ycol0 + n] = act( X[r,:D] @ W[:D, n0..n0+16) + bias ),  one 16x16 tile
// per wave, 4 waves per block. rows must be a multiple of 64, D of 4.
// ---------------------------------------------------------------------------
__global__ __launch_bounds__(128) void gemm16(
    const float* __restrict__ X, int D,
    const float* __restrict__ W, int ldW,
    const float* __restrict__ bias,
    float* __restrict__ Y, int ldY, int ycol0, int relu)
{
  const int wave = threadIdx.x >> 5;
  const int lane = threadIdx.x & 31;
  const int g    = lane >> 4;
  const int ln   = lane & 15;
  const long r0  = (long)(blockIdx.x * 4 + wave) * 16;
  const int  n0  = blockIdx.y * 16;

  v8f acc = {};
  const float* xrow = X + (r0 + ln) * (long)D;
  for (int k = 0; k < D; k += 4) {
    const int ka = k + 2 * g;
    v2f a, b;
    a[0] = xrow[ka];
    a[1] = xrow[ka + 1];
    b[0] = W[(long)ka * ldW + n0 + ln];
    b[1] = W[(long)(ka + 1) * ldW + n0 + ln];
    acc = __builtin_amdgcn_wmma_f32_16x16x4_f32(
        false, a, false, b, (short)0, acc, false, false);
  }
  const float bv = bias ? bias[n0 + ln] : 0.0f;
#pragma unroll
  for (int v = 0; v < 8; ++v) {
    float y = acc[v] + bv;
    if (relu) y = fmaxf(y, 0.0f);
    Y[(r0 + v + 8 * g) * (long)ldY + ycol0 + n0 + ln] = y;
  }
}

// ---------------------------------------------------------------------------
// Helpers: order-preserving float<->uint encoding for atomic max
// ---------------------------------------------------------------------------
__device__ __forceinline__ unsigned fenc(float f) {
  unsigned u = __float_as_uint(f);
  return (u >> 31) ? ~u : (u | 0x80000000u);
}
__device__ __forceinline__ float fdec(unsigned k) {
  unsigned u = (k >> 31) ? (k & 0x7FFFFFFFu) : ~k;
  return __uint_as_float(u);
}

__global__ void zero_kernel(unsigned* __restrict__ p, long n) {
  long i = blockIdx.x * (long)blockDim.x + threadIdx.x;
  if (i < n) p[i] = 0u;
}

// ---------------------------------------------------------------------------
// logits[e,rp,h] = (1/sqrt(K)) * sum_k qs[send,rp,h,k]*(qr[recv,rp,h,k]+ke[e,h,k])
// also atomic segment-max into nodeMax[recv,rp,h]
// ---------------------------------------------------------------------------
__global__ void logits_kernel(
    const float* __restrict__ qs, const float* __restrict__ qr,
    const float* __restrict__ ke,
    const int* __restrict__ senders, const int* __restrict__ receivers,
    float* __restrict__ wbuf, unsigned* __restrict__ nodeMax,
    int RP, long total)
{
  long tid = blockIdx.x * (long)blockDim.x + threadIdx.x;
  if (tid >= total) return;
  int h = (int)(tid & 3);
  long rem = tid >> 2;
  int rp = (int)(rem % RP);
  long e = rem / RP;
  long gg = e / E_;
  int sn = (int)(gg * N_) + senders[e];
  int rn = (int)(gg * N_) + receivers[e];
  const float4* ps = (const float4*)(qs + ((long)sn * RP + rp) * (HEADS * KK) + h * KK);
  const float4* pr = (const float4*)(qr + ((long)rn * RP + rp) * (HEADS * KK) + h * KK);
  const float4* pe = (const float4*)(ke + e * (HEADS * KK) + h * KK);
  float acc = 0.0f;
#pragma unroll
  for (int k = 0; k < 4; ++k) {
    float4 s = ps[k], r = pr[k], ev = pe[k];
    acc += s.x * (r.x + ev.x) + s.y * (r.y + ev.y) +
           s.z * (r.z + ev.z) + s.w * (r.w + ev.w);
  }
  acc *= 0.25f;  // 1/sqrt(16)
  wbuf[tid] = acc;
  atomicMax(&nodeMax[((long)rn * RP + rp) * HEADS + h], fenc(acc));
}

// wbuf <- exp(logit - segmax); atomic segment-sum into nodeSum
__global__ void expsum_kernel(
    const int* __restrict__ receivers,
    float* __restrict__ wbuf, const unsigned* __restrict__ nodeMax,
    float* __restrict__ nodeSum, int RP, long total)
{
  long tid = blockIdx.x * (long)blockDim.x + threadIdx.x;
  if (tid >= total) return;
  int h = (int)(tid & 3);
  long rem = tid >> 2;
  int rp = (int)(rem % RP);
  long e = rem / RP;
  long gg = e / E_;
  int rn = (int)(gg * N_) + receivers[e];
  long idxn = ((long)rn * RP + rp) * HEADS + h;
  float p = expf(wbuf[tid] - fdec(nodeMax[idxn]));
  wbuf[tid] = p;
  atomicAdd(&nodeSum[idxn], p);
}

// Scatter messages: out += w_norm * nh[send].
// One thread per (e, rp, c); loops over the 4 heads so the `sent` gather is
// read once and (layer 1) the head-mean collapses to a single atomic.
// mode 0 (layer 0): out = x1[node, r, h*32+c] with r = rp/3 (sums over P)
// mode 1 (layer 1): out = out1[node, rp, c], mean over heads
__global__ void scatter_kernel(
    const float* __restrict__ wbuf, const float* __restrict__ nodeSum,
    const float* __restrict__ nh,
    const int* __restrict__ senders, const int* __restrict__ receivers,
    float* __restrict__ out, int RP, int mode, long total)
{
  long tid = blockIdx.x * (long)blockDim.x + threadIdx.x;
  if (tid >= total) return;
  int c = (int)(tid & 31);
  long rem = tid >> 5;
  int rp = (int)(rem % RP);
  long e = rem / RP;
  long gg = e / E_;
  int sn = (int)(gg * N_) + senders[e];
  int rn = (int)(gg * N_) + receivers[e];
  float sent = nh[((long)sn * RP + rp) * HH + c];
  long wbase = (e * RP + rp) * HEADS;
  long nbase = ((long)rn * RP + rp) * HEADS;
  if (mode == 0) {
    int r = rp / 3;
    long dbase = ((long)rn * 4 + r) * (HEADS * HH) + c;
#pragma unroll
    for (int h = 0; h < HEADS; ++h) {
      float wn = wbuf[wbase + h] / nodeSum[nbase + h];
      atomicAdd(&out[dbase + h * HH], wn * sent);
    }
  } else {
    float acc = 0.0f;
#pragma unroll
    for (int h = 0; h < HEADS; ++h)
      acc += wbuf[wbase + h] / nodeSum[nbase + h];
    atomicAdd(&out[((long)rn * RP + rp) * HH + c], 0.25f * acc * sent);
  }
}

// y[node, c] = out1[node, r=3, c]   (head-mean already applied in scatter)
__global__ void final_kernel(const float* __restrict__ out1,
                             float* __restrict__ y)
{
  int i = blockIdx.x * blockDim.x + threadIdx.x;  // 131072 threads
  int node = i >> 5;
  int c = i & 31;
  y[i] = out1[((long)node * 4 + 3) * HH + c];
}

// ---------------------------------------------------------------------------
extern "C" void kernel_launch(void* const* d_in, const int* in_sizes, int n_in,
                              void* d_out, int out_size, void* d_ws, size_t ws_size,
                              hipStream_t stream) {
  (void)in_sizes; (void)n_in; (void)out_size; (void)ws_size;
  const float* eqn   = (const float*)d_in[0];  // (8,4,128,4,3,16)
  const float* edges = (const float*)d_in[1];  // (8,4,2048,8)
  const int* receivers = (const int*)d_in[66];
  const int* senders   = (const int*)d_in[67];

  // Workspace carve-up (floats)
  float* ws = (float*)d_ws;
  long off = 0;
  float* hid  = ws + off; off += 2097152;            // 65536x32 scratch
  float* nh   = ws + off; off += 1572864;            // node hidden (nh0 / nh1)
  float* eh   = ws + off; off += 2097152;            // edge hidden
  float* qs   = ws + off; off += 3145728;            // rows x (4*16)
  float* qr   = ws + off; off += 3145728;
  float* ke   = ws + off; off += 4194304;            // 65536 x 64
  float* wbuf = ws + off; off += 3145728;            // per-edge logits/weights
  unsigned* nodeMax = (unsigned*)(ws + off); off += 196608;
  float* nodeSum = ws + off; off += 196608;
  float* x1   = ws + off; off += 2097152;            // 16384 x 128
  float* out1 = ws + off; off += 524288;             // 4096 x 4 x 32

  for (int layer = 0; layer < 2; ++layer) {
    const int base = 2 + 32 * layer;
    // pytree order within a layer: edge_mlp(W1,b1,W2,b2), heads[4]{We,Wr,Ws,be,br,bs}, node_mlp(W1,b1,W2,b2)
    const float* eW1 = (const float*)d_in[base + 0];
    const float* eb1 = (const float*)d_in[base + 1];
    const float* eW2 = (const float*)d_in[base + 2];
    const float* eb2 = (const float*)d_in[base + 3];
    const float* nW1 = (const float*)d_in[base + 28];
    const float* nb1 = (const float*)d_in[base + 29];
    const float* nW2 = (const float*)d_in[base + 30];
    const float* nb2 = (const float*)d_in[base + 31];

    const int RP   = (layer == 0) ? RP0 : RP1;
    const long nrows = (long)NODES * RP;             // 49152 or 16384
    const int  D_in  = (layer == 0) ? 16 : 128;
    const float* Xn  = (layer == 0) ? eqn : x1;
    const int nrb = (int)(nrows / 64);               // row blocks (4 waves/blk)
    const int erb = EDG / 64;                        // 1024

    // Node MLP: D_in -> 32 (relu) -> 32 (relu)
    gemm16<<<dim3(nrb, 2), 128, 0, stream>>>(Xn, D_in, nW1, 32, nb1, hid, 32, 0, 1);
    gemm16<<<dim3(nrb, 2), 128, 0, stream>>>(hid, 32, nW2, 32, nb2, nh, 32, 0, 1);
    // Edge MLP: 8 -> 32 (relu) -> 32 (relu)
    gemm16<<<dim3(erb, 2), 128, 0, stream>>>(edges, 8, eW1, 32, eb1, hid, 32, 0, 1);
    gemm16<<<dim3(erb, 2), 128, 0, stream>>>(hid, 32, eW2, 32, eb2, eh, 32, 0, 1);

    // Per-head projections at node / edge level
    for (int h = 0; h < HEADS; ++h) {
      const float* We = (const float*)d_in[base + 4 + 6 * h + 0];
      const float* Wr = (const float*)d_in[base + 4 + 6 * h + 1];
      const float* Wsw = (const float*)d_in[base + 4 + 6 * h + 2];
      const float* be = (const float*)d_in[base + 4 + 6 * h + 3];
      const float* br = (const float*)d_in[base + 4 + 6 * h + 4];
      const float* bs = (const float*)d_in[base + 4 + 6 * h + 5];
      gemm16<<<dim3(nrb, 1), 128, 0, stream>>>(nh, 32, Wsw, 16, bs, qs, 64, h * 16, 0);
      gemm16<<<dim3(nrb, 1), 128, 0, stream>>>(nh, 32, Wr, 16, br, qr, 64, h * 16, 0);
      gemm16<<<dim3(erb, 1), 128, 0, stream>>>(eh, 32, We, 16, be, ke, 64, h * 16, 0);
    }

    // Zero segment-max/sum and the output accumulator
    zero_kernel<<<(196608 + 255) / 256, 256, 0, stream>>>(nodeMax, 196608);
    zero_kernel<<<(196608 + 255) / 256, 256, 0, stream>>>((unsigned*)nodeSum, 196608);
    if (layer == 0)
      zero_kernel<<<(2097152 + 255) / 256, 256, 0, stream>>>((unsigned*)x1, 2097152);
    else
      zero_kernel<<<(524288 + 255) / 256, 256, 0, stream>>>((unsigned*)out1, 524288);

    // Attention: logits + segmax, exp + segsum, normalized scatter
    const long tot = (long)EDG * RP * HEADS;
    logits_kernel<<<(unsigned)((tot + 255) / 256), 256, 0, stream>>>(
        qs, qr, ke, senders, receivers, wbuf, nodeMax, RP, tot);
    expsum_kernel<<<(unsigned)((tot + 255) / 256), 256, 0, stream>>>(
        receivers, wbuf, nodeMax, nodeSum, RP, tot);
    const long totc = (long)EDG * RP * HH;
    scatter_kernel<<<(unsigned)((totc + 255) / 256), 256, 0, stream>>>(
        wbuf, nodeSum, nh, senders, receivers,
        (layer == 0) ? x1 : out1, RP, layer, totc);
  }

  final_kernel<<<(NODES * HH) / 256, 256, 0, stream>>>(out1, (float*)d_out);
}